// GCNBlock_41188736368774
// MI455X (gfx1250) — compile-verified
//
#include <hip/hip_runtime.h>
#include <hip/hip_bf16.h>

#define N_NODES 50000
#define N_EDGES 600000
#define CHANNELS 128
#define LN_EPS 1e-5f
#define NEG_SLOPE 0.01f

typedef __attribute__((ext_vector_type(2))) float v2f;
typedef __attribute__((ext_vector_type(8))) float v8f;

// ---------------- Kernel 1: deg init (self loop = 1.0) ----------------
__global__ void k_deg_init(float* __restrict__ deg) {
    int n = blockIdx.x * blockDim.x + threadIdx.x;
    if (n < N_NODES) deg[n] = 1.0f;
}

// ---------------- Kernel 2: degree accumulation over dst ----------------
__global__ void k_deg_accum(const int* __restrict__ dst, float* __restrict__ deg) {
    int e = blockIdx.x * blockDim.x + threadIdx.x;
    if (e < N_EDGES) {
        atomicAdd(&deg[dst[e]], 1.0f);
    }
}

// ---------------- Kernel 3: dinv = rsqrt(deg), in place ----------------
__global__ void k_dinv(float* __restrict__ deg) {
    int n = blockIdx.x * blockDim.x + threadIdx.x;
    if (n < N_NODES) deg[n] = __frsqrt_rn(deg[n]);
}

// ---------------- Kernel 4: xw = x @ W via V_WMMA_F32_16X16X4_F32 ----------
// One wave computes one 16x16 tile of xw. Block = 256 threads = 8 waves.
// blockIdx.x = M-tile (3125 tiles of 16 rows); wave id = N-tile (8 tiles of 16 cols).
// A (16x4) layout: lane<16 -> M=lane, K={k,k+1}; lane>=16 -> M=lane-16, K={k+2,k+3}.
// B (4x16) layout: v0 lanes0-15 = row k (N=lane), lanes16-31 = row k+2; v1 = k+1 / k+3.
// C/D (16x16): c[v] at row v + 8*(lane>=16), col = lane&15.
__global__ void __launch_bounds__(256)
k_gemm_wmma(const float* __restrict__ x, const float* __restrict__ W,
            float* __restrict__ xw) {
    const int lane = threadIdx.x & 31;
    const int wave = threadIdx.x >> 5;          // 0..7 -> N tile
    const int rm   = blockIdx.x << 4;           // row base
    const int cn   = wave << 4;                 // col base

    const int ml = lane & 15;                   // M (for A) / N (for B/C)
    const int kh = (lane >> 4) << 1;            // 0 or 2: K sub-offset
    const int hi = (lane >> 4) << 3;            // 0 or 8: C/D row half

    const float* arow = x + (size_t)(rm + ml) * CHANNELS + kh;
    const float* bcol = W + (size_t)kh * CHANNELS + cn + ml;

    v8f c = {0.f, 0.f, 0.f, 0.f, 0.f, 0.f, 0.f, 0.f};

    #pragma unroll
    for (int k = 0; k < CHANNELS; k += 4) {
        v2f a, b;
        a.x = arow[k + 0];
        a.y = arow[k + 1];
        b.x = bcol[(size_t)k * CHANNELS];
        b.y = bcol[(size_t)(k + 1) * CHANNELS];
        // 8 args: (neg_a, A, neg_b, B, c_mod, C, reuse_a, reuse_b)
        c = __builtin_amdgcn_wmma_f32_16x16x4_f32(false, a, false, b,
                                                  (short)0, c, false, false);
    }

    float* orow = xw + (size_t)(rm + hi) * CHANNELS + cn + ml;
    #pragma unroll
    for (int v = 0; v < 8; ++v) {
        orow[(size_t)v * CHANNELS] = c[v];
    }
}

// ------- Kernel 5: agg = xw * dinv^2 + b   (self-loop contribution + bias) ----
// One wave per node row, each lane handles 4 channels (float4).
__global__ void k_agg_init(const float* __restrict__ xw, const float* __restrict__ dinv,
                           const float* __restrict__ b, float* __restrict__ agg) {
    int t = blockIdx.x * blockDim.x + threadIdx.x;
    int n = t >> 5;
    int lane = t & 31;
    if (n >= N_NODES) return;
    float di = dinv[n];
    float s = di * di;
    int c0 = lane << 2;
    const float4 v = *(const float4*)(xw + (size_t)n * CHANNELS + c0);
    const float4 bb = *(const float4*)(b + c0);
    float4 o;
    o.x = v.x * s + bb.x;
    o.y = v.y * s + bb.y;
    o.z = v.z * s + bb.z;
    o.w = v.w * s + bb.w;
    *(float4*)(agg + (size_t)n * CHANNELS + c0) = o;
}

// ------- Kernel 6: edge scatter:  agg[dst] += xw[src] * dinv[src]*dinv[dst] ----
// 32 lanes per edge; each lane gathers a float4 and issues 4 f32 atomics.
__global__ void k_scatter(const int* __restrict__ src, const int* __restrict__ dst,
                          const float* __restrict__ dinv,
                          const float* __restrict__ xw, float* __restrict__ agg) {
    long long t = (long long)blockIdx.x * blockDim.x + threadIdx.x;
    int e = (int)(t >> 5);
    int lane = (int)(t & 31);
    if (e >= N_EDGES) return;
    int s = src[e];
    int d = dst[e];
    float nr = dinv[s] * dinv[d];
    int c0 = lane << 2;
    const float4 v = *(const float4*)(xw + (size_t)s * CHANNELS + c0);
    float* ag = agg + (size_t)d * CHANNELS + c0;
    atomicAdd(ag + 0, v.x * nr);
    atomicAdd(ag + 1, v.y * nr);
    atomicAdd(ag + 2, v.z * nr);
    atomicAdd(ag + 3, v.w * nr);
}

// ------- Kernel 7: LayerNorm + LeakyReLU + residual ----------------------
__device__ __forceinline__ float wave_sum32(float v) {
    #pragma unroll
    for (int off = 16; off >= 1; off >>= 1) v += __shfl_xor(v, off, 32);
    return v;
}

__global__ void k_ln_act_res(const float* __restrict__ agg, const float* __restrict__ x,
                             const float* __restrict__ gamma, const float* __restrict__ beta,
                             float* __restrict__ out) {
    int t = blockIdx.x * blockDim.x + threadIdx.x;
    int n = t >> 5;
    int lane = t & 31;
    if (n >= N_NODES) return;
    int c0 = lane << 2;
    const float4 v = *(const float4*)(agg + (size_t)n * CHANNELS + c0);

    float s = v.x + v.y + v.z + v.w;
    float mu = wave_sum32(s) * (1.0f / CHANNELS);

    float dx = v.x - mu, dy = v.y - mu, dz = v.z - mu, dw = v.w - mu;
    float s2 = dx * dx + dy * dy + dz * dz + dw * dw;
    float var = wave_sum32(s2) * (1.0f / CHANNELS);
    float inv = __frsqrt_rn(var + LN_EPS);

    const float4 g = *(const float4*)(gamma + c0);
    const float4 be = *(const float4*)(beta + c0);
    const float4 xr = *(const float4*)(x + (size_t)n * CHANNELS + c0);

    float4 o;
    o.x = dx * inv * g.x + be.x;
    o.y = dy * inv * g.y + be.y;
    o.z = dz * inv * g.z + be.z;
    o.w = dw * inv * g.w + be.w;
    o.x = (o.x >= 0.f ? o.x : NEG_SLOPE * o.x) + xr.x;
    o.y = (o.y >= 0.f ? o.y : NEG_SLOPE * o.y) + xr.y;
    o.z = (o.z >= 0.f ? o.z : NEG_SLOPE * o.z) + xr.z;
    o.w = (o.w >= 0.f ? o.w : NEG_SLOPE * o.w) + xr.w;
    *(float4*)(out + (size_t)n * CHANNELS + c0) = o;
}

extern "C" void kernel_launch(void* const* d_in, const int* in_sizes, int n_in,
                              void* d_out, int out_size, void* d_ws, size_t ws_size,
                              hipStream_t stream) {
    const float* x     = (const float*)d_in[0];
    const int*   eidx  = (const int*)d_in[1];   // [2, E]
    const float* W     = (const float*)d_in[2];
    const float* b     = (const float*)d_in[3];
    const float* gamma = (const float*)d_in[4];
    const float* beta  = (const float*)d_in[5];
    float* out = (float*)d_out;

    const int* src = eidx;             // row 0
    const int* dst = eidx + N_EDGES;   // row 1

    // workspace layout (256B aligned):
    //   deg/dinv : N floats
    //   xw       : N*C floats
    //   agg      : N*C floats
    char* ws = (char*)d_ws;
    size_t off = 0;
    float* deg = (float*)(ws + off);
    off += ((size_t)N_NODES * 4 + 255) & ~(size_t)255;
    float* xw = (float*)(ws + off);
    off += ((size_t)N_NODES * CHANNELS * 4 + 255) & ~(size_t)255;
    float* agg = (float*)(ws + off);

    // 1. deg = 1 (self loop)
    k_deg_init<<<(N_NODES + 255) / 256, 256, 0, stream>>>(deg);
    // 2. deg += per-dst edge count
    k_deg_accum<<<(N_EDGES + 255) / 256, 256, 0, stream>>>(dst, deg);
    // 3. dinv = rsqrt(deg)
    k_dinv<<<(N_NODES + 255) / 256, 256, 0, stream>>>(deg);
    // 4. xw = x @ W (WMMA f32 16x16x4), 3125 M-tiles x 8 N-tiles, 8 waves/block
    k_gemm_wmma<<<N_NODES / 16, 256, 0, stream>>>(x, W, xw);
    // 5. agg = xw*dinv^2 + b
    k_agg_init<<<(N_NODES * 32 + 255) / 256, 256, 0, stream>>>(xw, deg, b, agg);
    // 6. scatter-add messages
    {
        long long threads = (long long)N_EDGES * 32;
        int blocks = (int)((threads + 255) / 256);
        k_scatter<<<blocks, 256, 0, stream>>>(src, dst, deg, xw, agg);
    }
    // 7. LayerNorm + LeakyReLU + residual
    k_ln_act_res<<<(N_NODES * 32 + 255) / 256, 256, 0, stream>>>(agg, x, gamma, beta, out);
}